// DCT_30382598651944
// MI455X (gfx1250) — compile-verified
//
#include <hip/hip_runtime.h>
#include <hip/hip_bf16.h>

typedef __attribute__((ext_vector_type(2))) float v2f;
typedef __attribute__((ext_vector_type(8))) float v8f;

// zigzag order for 8x8 (position -> row*8+col), standard JPEG order
__constant__ int c_zz[64] = {
     0, 1, 8,16, 9, 2, 3,10,
    17,24,32,25,18,11, 4, 5,
    12,19,26,33,40,48,41,34,
    27,20,13, 6, 7,14,21,28,
    35,42,49,56,57,50,43,36,
    29,22,15,23,30,37,44,51,
    58,59,52,45,38,31,39,46,
    53,60,61,54,47,55,62,63};

// cos(k*pi/16), k = 0..8
__constant__ float c_cos[9] = {
    1.0f,
    0.98078528040323044913f,
    0.92387953251128675613f,
    0.83146961230254523708f,
    0.70710678118654752440f,
    0.55557023301960222474f,
    0.38268343236508977173f,
    0.19509032201612826785f,
    0.0f};

#define Bd 4
#define Cd 3
#define Td 64
#define Hd 224
#define Wd 224
#define NBH 28        // block rows per frame
#define NT4 7         // 32-column tiles per row (4 blocks each)
#define HW (Hd*Wd)
#define CH_STRIDE (Td*Hd*Wd)

// DCT basis: D[l][n] = 2*cos(pi*(2n+1)*l/16) via table + symmetry
__device__ __forceinline__ float dmat(int l, int n) {
    int u = ((2 * n + 1) * l) & 31;          // angle in pi/16 units mod 2*pi
    if (u > 16) u = 32 - u;                  // cos(-x) = cos(x)
    float s = 2.0f;
    int idx = u;
    if (u > 8) { idx = 16 - u; s = -2.0f; }  // cos(pi - x) = -cos(x)
    return s * c_cos[idx];
}

__global__ __launch_bounds__(32) void dct_zz_kernel(const float* __restrict__ x,
                                                    const int* __restrict__ remp,
                                                    float* __restrict__ out) {
    __shared__ float smem[256];   // 4 blocks x 64 coeffs (single-wave workgroup)

    const int lane = threadIdx.x;
    const int hi   = lane >> 4;   // half-wave id
    const int N    = lane & 15;   // column (B/C/D) or row M (A) index
    const int row  = N & 7;       // row within an 8x8 block
    const int rblk = N >> 3;      // which block of a pair owns this A-row

    const int keep  = 63 - remp[0];   // 48 coefficients kept per block
    const int keep4 = 4 * keep;       // outputs per 4-block tile

    // workgroup = one block-row: (b, t, hb)
    int gw = blockIdx.x;
    const int hb = gw % NBH;  gw /= NBH;
    const int t  = gw % Td;   gw /= Td;
    const int b  = gw;

    // ---- constant operand fragments (built once per wave) ----
    // B1[n][l] = D[l][n] (l<8), zero-padded cols 8..15. b1[s][v] = B1[4s+2hi+v][N]
    v2f b1[2];
    #pragma unroll
    for (int s = 0; s < 2; ++s)
        #pragma unroll
        for (int v = 0; v < 2; ++v)
            b1[s][v] = (N < 8) ? dmat(N, 4*s + 2*hi + v) : 0.0f;
    // A2 = diag(D,D): a2[s][v] = D2[M=N][K=4s+2hi+v], K>>3 == s>>1
    v2f a2[4];
    #pragma unroll
    for (int s = 0; s < 4; ++s)
        #pragma unroll
        for (int v = 0; v < 2; ++v)
            a2[s][v] = ((N >> 3) == (s >> 1)) ? dmat(row, (4*s + 2*hi + v) & 7) : 0.0f;

    // ---- per-slot zigzag gather precompute (blk in 0..3 within 4-block tile) ----
    int lds_idx[8];
    #pragma unroll
    for (int s = 0; s < 8; ++s) {
        const int u = lane + 32 * s;
        if (u < keep4) {
            const int blk = (u >= keep) + (u >= 2*keep) + (u >= 3*keep);
            const int j   = u - blk * keep;
            lds_idx[s] = blk * 64 + c_zz[1 + j];
        } else {
            lds_idx[s] = 0;
        }
    }

    // ---- base addresses ----
    long ubase = ((long)(b * Cd) * Td + t) * (long)HW + (long)(hb * 8) * Wd;
    const int loff = row * Wd + rblk * 8 + 2 * hi;   // per-lane offset (c=0)
    long obase = ((long)(b * Td + t) * (NBH * NBH) + hb * NBH) * keep;  // 784 = 28*28

    const float wc0 = 0.2989f, wc1 = 0.587f, wc2 = 0.114f;

    for (int q2 = 0; q2 < NT4; ++q2) {
        // ---- issue all 12 NT loads up front (one full 128B line per row) ----
        const float* base = x + ubase + loff;
        v2f L[12];
        #pragma unroll
        for (int e = 0; e < 2; ++e) {
            const float* be = base + e * 16;
            L[e*6 + 0] = __builtin_nontemporal_load((const v2f*)(be));
            L[e*6 + 1] = __builtin_nontemporal_load((const v2f*)(be + 4));
            L[e*6 + 2] = __builtin_nontemporal_load((const v2f*)(be + (long)CH_STRIDE));
            L[e*6 + 3] = __builtin_nontemporal_load((const v2f*)(be + (long)CH_STRIDE + 4));
            L[e*6 + 4] = __builtin_nontemporal_load((const v2f*)(be + 2L * CH_STRIDE));
            L[e*6 + 5] = __builtin_nontemporal_load((const v2f*)(be + 2L * CH_STRIDE + 4));
        }

        v8f c2e[2];
        #pragma unroll
        for (int e = 0; e < 2; ++e) {
            v2f gA = wc0 * L[e*6 + 0] + wc1 * L[e*6 + 2] + wc2 * L[e*6 + 4];
            v2f gB = wc0 * L[e*6 + 1] + wc1 * L[e*6 + 3] + wc2 * L[e*6 + 5];

            // stage 1: T1 = [g0;g1] * D^T  (K=8 -> 2 WMMA)
            v8f c1 = {};
            c1 = __builtin_amdgcn_wmma_f32_16x16x4_f32(false, gA, false, b1[0], (short)0, c1, false, false);
            c1 = __builtin_amdgcn_wmma_f32_16x16x4_f32(false, gB, false, b1[1], (short)0, c1, false, false);

            // repack C/D-layout c1 into B-operand fragments via lane^16 exchange
            v2f b2[4];
            #pragma unroll
            for (int s = 0; s < 4; ++s) {
                #pragma unroll
                for (int v = 0; v < 2; ++v) {
                    const int ia = (4*s + v) & 7;
                    const int ib = (4*s + 2 + v) & 7;
                    float send  = hi ? c1[ia] : c1[ib];
                    float keepv = hi ? c1[ib] : c1[ia];
                    float recv  = __shfl_xor(send, 16, 32);
                    b2[s][v] = ((s >> 1) == hi) ? keepv : recv;
                }
            }

            // stage 2: OUT = diag(D,D) * T1  (K=16 -> 4 WMMA)
            v8f c2 = {};
            #pragma unroll
            for (int s = 0; s < 4; ++s)
                c2 = __builtin_amdgcn_wmma_f32_16x16x4_f32(false, a2[s], false, b2[s], (short)0, c2, false, false);
            c2e[e] = c2;
        }

        // ---- scatter all 4 blocks' coefficients to LDS (single branch) ----
        if (N < 8) {
            #pragma unroll
            for (int e = 0; e < 2; ++e)
                #pragma unroll
                for (int r = 0; r < 8; ++r)
                    smem[(e * 2 + hi) * 64 + r * 8 + N] = c2e[e][r];
        }
        __syncthreads();   // single-wave WG: HW-NOP barrier, keeps DS ordering

        // ---- zigzag gather + coalesced NT stores (192 contiguous floats) ----
        #pragma unroll
        for (int s = 0; s < 8; ++s) {
            const int u = lane + 32 * s;
            if (u < keep4) {
                const float val = smem[lds_idx[s]];
                __builtin_nontemporal_store(val, out + obase + u);
            }
        }
        __syncthreads();   // protect smem reuse next iteration

        ubase += 32;       // next tile: +32 columns
        obase += keep4;    // next tile: +4 blocks of keep coeffs
    }
}

extern "C" void kernel_launch(void* const* d_in, const int* in_sizes, int n_in,
                              void* d_out, int out_size, void* d_ws, size_t ws_size,
                              hipStream_t stream) {
    const float* x  = (const float*)d_in[0];
    const int* rem  = (const int*)d_in[1];
    float* out      = (float*)d_out;

    // one single-wave workgroup per (b, t, hb) block-row: 4*64*28 = 7168
    const int blocks = Bd * Td * NBH;
    dct_zz_kernel<<<blocks, 32, 0, stream>>>(x, rem, out);
}